// ADMMBlock_72224170050161
// MI455X (gfx1250) — compile-verified
//
#include <hip/hip_runtime.h>
#include <math.h>

// Problem constants (match reference)
#define Bdim 2
#define Tdim 12
#define Ndim 2500
#define Hdim 4
#define Cdim 4
#define Kdim 8
#define NADMM 2
#define NCG 2

#define NV4   (Bdim*Tdim*Ndim*Hdim)   // 240000 float4 work items per tensor
#define NELEM (NV4*Cdim)              // 960000 floats per tensor
#define MROWS (Bdim*Tdim*Ndim)        // 60000 rows of 16 (h,c) floats

static_assert(MROWS % 16 == 0, "combine kernel exploits 16-row alignment");

typedef __attribute__((ext_vector_type(2))) float v2f;
typedef __attribute__((ext_vector_type(8))) float v8f;

// ---------- small helpers ----------
__device__ __forceinline__ float4 ld4(const float* p) {
  return *reinterpret_cast<const float4*>(p);
}
__device__ __forceinline__ void st4(float* p, float4 v) {
  *reinterpret_cast<float4*>(p) = v;
}
__device__ __forceinline__ float4 mk4(float a, float b, float c, float d) {
  float4 r; r.x = a; r.y = b; r.z = c; r.w = d; return r;
}
__device__ __forceinline__ void decomp(int tid, int& b, int& t, int& n, int& h) {
  h = tid & (Hdim - 1);
  int r = tid >> 2;          // Hdim == 4
  n = r % Ndim;
  r /= Ndim;
  t = r % Tdim;
  b = r / Tdim;
}

// ldr(v)[b,t,n,h,:] for t>=1 : v[t] - sum_k d_ew[n,t-1,k,h] * v[t-1, knn[n,k]]
__device__ __forceinline__ float4 gather_ldr(const float* __restrict__ v,
                                             const float* __restrict__ d_ew,
                                             const int* __restrict__ knn,
                                             int b, int t, int n, int h) {
  float ax = 0.f, ay = 0.f, az = 0.f, aw = 0.f;
  const int* kr = knn + n * Kdim;
  const float* wrow = d_ew + ((size_t)(n * (Tdim - 1) + (t - 1)) * Kdim) * Hdim + h;
  const int tb = (b * Tdim + (t - 1)) * Ndim;
#pragma unroll
  for (int k = 0; k < Kdim; ++k) {
    float wk = wrow[k * Hdim];
    int nn = kr[k];
    float4 xv = ld4(v + 4 * ((tb + nn) * Hdim + h));
    ax += wk * xv.x; ay += wk * xv.y; az += wk * xv.z; aw += wk * xv.w;
  }
  float4 vc = ld4(v + 4 * (((b * Tdim + t) * Ndim + n) * Hdim + h));
  return mk4(vc.x - ax, vc.y - ay, vc.z - az, vc.w - aw);
}

// ldr_t(v)[b,t,n,h,:] : t==T-1 -> 0 ; agg = sum_k d_ew[n,t,k,h]*v[t+1,knn[n,k]]
//                       t==0 -> -agg ; else v[t]-agg
__device__ __forceinline__ float4 gather_ldrt(const float* __restrict__ v,
                                              const float* __restrict__ d_ew,
                                              const int* __restrict__ knn,
                                              int b, int t, int n, int h) {
  if (t == Tdim - 1) return mk4(0.f, 0.f, 0.f, 0.f);
  float ax = 0.f, ay = 0.f, az = 0.f, aw = 0.f;
  const int* kr = knn + n * Kdim;
  const float* wrow = d_ew + ((size_t)(n * (Tdim - 1) + t) * Kdim) * Hdim + h;
  const int tb = (b * Tdim + (t + 1)) * Ndim;
#pragma unroll
  for (int k = 0; k < Kdim; ++k) {
    float wk = wrow[k * Hdim];
    int nn = kr[k];
    float4 xv = ld4(v + 4 * ((tb + nn) * Hdim + h));
    ax += wk * xv.x; ay += wk * xv.y; az += wk * xv.z; aw += wk * xv.w;
  }
  if (t >= 1) {
    float4 vc = ld4(v + 4 * (((b * Tdim + t) * Ndim + n) * Hdim + h));
    return mk4(vc.x - ax, vc.y - ay, vc.z - az, vc.w - aw);
  }
  return mk4(-ax, -ay, -az, -aw);
}

#define TID_GUARD int tid = blockIdx.x * blockDim.x + threadIdx.x; if (tid >= NV4) return;

// ---------- kernels ----------
__global__ void k_init(float* __restrict__ x, float* __restrict__ zu, float* __restrict__ zd,
                       float* __restrict__ g, float* __restrict__ gu, float* __restrict__ gd,
                       const float* __restrict__ y) {
  TID_GUARD
  int b, t, n, h; decomp(tid, b, t, n, h); (void)h;
  float4 yv = ld4(y + 4 * ((b * Tdim + t) * Ndim + n));
  st4(x + 4 * tid, yv); st4(zu + 4 * tid, yv); st4(zd + 4 * tid, yv);
  float4 g0 = mk4(0.1f, 0.1f, 0.1f, 0.1f);
  st4(g + 4 * tid, g0); st4(gu + 4 * tid, g0); st4(gd + 4 * tid, g0);
}

__global__ void k_ldr(float* __restrict__ out, const float* __restrict__ v,
                      const float* __restrict__ d_ew, const int* __restrict__ knn) {
  TID_GUARD
  int b, t, n, h; decomp(tid, b, t, n, h);
  float4 o = (t == 0) ? mk4(0.f, 0.f, 0.f, 0.f) : gather_ldr(v, d_ew, knn, b, t, n, h);
  st4(out + 4 * tid, o);
}

// out = (mu_u + rho_u/2)*v - mu_u * sum_k u_ew*v_gathered   (== mu_u*lu(v)+rho_u/2*v)
__global__ void k_lu(float* __restrict__ out, const float* __restrict__ v,
                     const float* __restrict__ u_ew, const int* __restrict__ knn,
                     const float* __restrict__ mu_u, const float* __restrict__ rho_u, int i) {
  TID_GUARD
  int b, t, n, h; decomp(tid, b, t, n, h);
  float mu = mu_u[i];
  float cv = mu + 0.5f * rho_u[i];
  float ax = 0.f, ay = 0.f, az = 0.f, aw = 0.f;
  const int* kr = knn + n * Kdim;
  const float* wrow = u_ew + ((size_t)(n * Tdim + t) * Kdim) * Hdim + h;
  const int tb = (b * Tdim + t) * Ndim;
#pragma unroll
  for (int k = 0; k < Kdim; ++k) {
    float wk = wrow[k * Hdim];
    int nn = kr[k];
    float4 xv = ld4(v + 4 * ((tb + nn) * Hdim + h));
    ax += wk * xv.x; ay += wk * xv.y; az += wk * xv.z; aw += wk * xv.w;
  }
  float4 vc = ld4(v + 4 * tid);
  st4(out + 4 * tid, mk4(cv * vc.x - mu * ax, cv * vc.y - mu * ay,
                         cv * vc.z - mu * az, cv * vc.w - mu * aw));
}

// Ap = rho/2 * ldrt(t1) + (rho_u/2 + rho_d/2 + [t<mask]) * v
__global__ void k_lhsx2(float* __restrict__ Ap, const float* __restrict__ t1,
                        const float* __restrict__ v, const float* __restrict__ d_ew,
                        const int* __restrict__ knn, const float* __restrict__ rho,
                        const float* __restrict__ rho_u, const float* __restrict__ rho_d,
                        int i, const int* __restrict__ maskp) {
  TID_GUARD
  int b, t, n, h; decomp(tid, b, t, n, h);
  float rr = 0.5f * rho[i];
  float cv = 0.5f * (rho_u[i] + rho_d[i]) + ((t < maskp[0]) ? 1.0f : 0.0f);
  float4 L = gather_ldrt(t1, d_ew, knn, b, t, n, h);
  float4 vc = ld4(v + 4 * tid);
  st4(Ap + 4 * tid, mk4(rr * L.x + cv * vc.x, rr * L.y + cv * vc.y,
                        rr * L.z + cv * vc.z, rr * L.w + cv * vc.w));
}

// out = mu_d2 * ldrt(t1) + rho_d/2 * v
__global__ void k_lhszd2(float* __restrict__ out, const float* __restrict__ t1,
                         const float* __restrict__ v, const float* __restrict__ d_ew,
                         const int* __restrict__ knn, const float* __restrict__ mu_d2,
                         const float* __restrict__ rho_d, int i) {
  TID_GUARD
  int b, t, n, h; decomp(tid, b, t, n, h);
  float cm = mu_d2[i];
  float cv = 0.5f * rho_d[i];
  float4 L = gather_ldrt(t1, d_ew, knn, b, t, n, h);
  float4 vc = ld4(v + 4 * tid);
  st4(out + 4 * tid, mk4(cm * L.x + cv * vc.x, cm * L.y + cv * vc.y,
                         cm * L.z + cv * vc.z, cm * L.w + cv * vc.w));
}

// rhs = 0.5 * ldrt(t1) + t2
__global__ void k_rhsx2(float* __restrict__ rhs, const float* __restrict__ t1,
                        const float* __restrict__ t2, const float* __restrict__ d_ew,
                        const int* __restrict__ knn) {
  TID_GUARD
  int b, t, n, h; decomp(tid, b, t, n, h);
  float4 L = gather_ldrt(t1, d_ew, knn, b, t, n, h);
  float4 o = ld4(t2 + 4 * tid);
  st4(rhs + 4 * tid, mk4(0.5f * L.x + o.x, 0.5f * L.y + o.y,
                         0.5f * L.z + o.z, 0.5f * L.w + o.w));
}

// t1 = gamma + rho*phi
__global__ void k_t1(float* __restrict__ out, const float* __restrict__ g,
                     const float* __restrict__ phi, const float* __restrict__ rho, int i) {
  TID_GUARD
  float rr = rho[i];
  float4 gv = ld4(g + 4 * tid);
  float4 pv = ld4(phi + 4 * tid);
  st4(out + 4 * tid, mk4(gv.x + rr * pv.x, gv.y + rr * pv.y,
                         gv.z + rr * pv.z, gv.w + rr * pv.w));
}

// t2 = rho_u/2*zu + rho_d/2*zd - (gu+gd)/2 + Hty
__global__ void k_rhspre(float* __restrict__ out, const float* __restrict__ zu,
                         const float* __restrict__ zd, const float* __restrict__ gu,
                         const float* __restrict__ gd, const float* __restrict__ y,
                         const float* __restrict__ rho_u, const float* __restrict__ rho_d,
                         int i, const int* __restrict__ maskp) {
  TID_GUARD
  int b, t, n, h; decomp(tid, b, t, n, h); (void)h;
  float ru = 0.5f * rho_u[i], rd = 0.5f * rho_d[i];
  float4 a = ld4(zu + 4 * tid), bb = ld4(zd + 4 * tid);
  float4 c = ld4(gu + 4 * tid), d = ld4(gd + 4 * tid);
  float4 o = mk4(ru * a.x + rd * bb.x - 0.5f * (c.x + d.x),
                 ru * a.y + rd * bb.y - 0.5f * (c.y + d.y),
                 ru * a.z + rd * bb.z - 0.5f * (c.z + d.z),
                 ru * a.w + rd * bb.w - 0.5f * (c.w + d.w));
  if (t < maskp[0]) {
    float4 yv = ld4(y + 4 * ((b * Tdim + t) * Ndim + n));
    o.x += yv.x; o.y += yv.y; o.z += yv.z; o.w += yv.w;
  }
  st4(out + 4 * tid, o);
}

// out = 0.5*(g + rho*x)
__global__ void k_halfaxpy(float* __restrict__ out, const float* __restrict__ g,
                           const float* __restrict__ x, const float* __restrict__ rho, int i) {
  TID_GUARD
  float rr = rho[i];
  float4 gv = ld4(g + 4 * tid), xv = ld4(x + 4 * tid);
  st4(out + 4 * tid, mk4(0.5f * (gv.x + rr * xv.x), 0.5f * (gv.y + rr * xv.y),
                         0.5f * (gv.z + rr * xv.z), 0.5f * (gv.w + rr * xv.w)));
}

// r = rhs - Ap ; p = r
__global__ void k_residual(float* __restrict__ r, float* __restrict__ p,
                           const float* __restrict__ rhs, const float* __restrict__ Ap) {
  TID_GUARD
  float4 a = ld4(rhs + 4 * tid), b = ld4(Ap + 4 * tid);
  float4 o = mk4(a.x - b.x, a.y - b.y, a.z - b.z, a.w - b.w);
  st4(r + 4 * tid, o); st4(p + 4 * tid, o);
}

// x += al[h]*p ; r -= al[h]*Ap ; p = r_new + be[h]*p_old
__global__ void k_cgupdate(float* __restrict__ x0, float* __restrict__ r, float* __restrict__ p,
                           const float* __restrict__ Ap, const float* __restrict__ al,
                           const float* __restrict__ be, int off) {
  TID_GUARD
  int h = tid & (Hdim - 1);
  float a = al[off + h], bc = be[off + h];
  float4 xv = ld4(x0 + 4 * tid), rv = ld4(r + 4 * tid);
  float4 pv = ld4(p + 4 * tid),  av = ld4(Ap + 4 * tid);
  xv.x += a * pv.x; xv.y += a * pv.y; xv.z += a * pv.z; xv.w += a * pv.w;
  rv.x -= a * av.x; rv.y -= a * av.y; rv.z -= a * av.z; rv.w -= a * av.w;
  float4 pn = mk4(rv.x + bc * pv.x, rv.y + bc * pv.y, rv.z + bc * pv.z, rv.w + bc * pv.w);
  st4(x0 + 4 * tid, xv); st4(r + 4 * tid, rv); st4(p + 4 * tid, pn);
}

// g += rho*(x - z)
__global__ void k_gammaupd(float* __restrict__ g, const float* __restrict__ x,
                           const float* __restrict__ z, const float* __restrict__ rho, int i) {
  TID_GUARD
  float rr = rho[i];
  float4 gv = ld4(g + 4 * tid), xv = ld4(x + 4 * tid), zv = ld4(z + 4 * tid);
  st4(g + 4 * tid, mk4(gv.x + rr * (xv.x - zv.x), gv.y + rr * (xv.y - zv.y),
                       gv.z + rr * (xv.z - zv.z), gv.w + rr * (xv.w - zv.w)));
}

// s = ldx - gamma/rho ; u = |s|-mu_d1/rho ; phi = sign(s)*max(u,0) ; gamma += rho*(phi-ldx)
__global__ void k_shrink(float* __restrict__ phi, float* __restrict__ gamma,
                         const float* __restrict__ ldx, const float* __restrict__ rho,
                         const float* __restrict__ mu_d1, int i) {
  TID_GUARD
  float rr = rho[i];
  float inv = 1.0f / rr;
  float th = mu_d1[i] * inv;
  float4 lv = ld4(ldx + 4 * tid);
  float4 gv = ld4(gamma + 4 * tid);
  float4 pv, gn;
#define SHRINK1(c) { float s = lv.c - gv.c * inv; float u = fabsf(s) - th; \
    float ph = (u > 0.f) ? ((s > 0.f) ? u : -u) : 0.f; \
    pv.c = ph; gn.c = gv.c + rr * (ph - lv.c); }
  SHRINK1(x) SHRINK1(y) SHRINK1(z) SHRINK1(w)
#undef SHRINK1
  st4(phi + 4 * tid, pv); st4(gamma + 4 * tid, gn);
}

// Final combine out[m,c] = sum_h w[h]*x[m, h*4+c] expressed as D = A(16x16) * B(16x4)
// via 4 accumulating V_WMMA_F32_16X16X4_F32 ops (K=4 chunks over j=(h,c)).
// B[j,n] = w[j>>2] * (n == (j&3)). Full fp32 precision.
// Mtot is a multiple of 16, so the only divergence is the post-WMMA store mask
// (lanes with N<4 hold the valid output columns); EXEC is all-ones through WMMAs.
__global__ void k_combine_wmma(const float* __restrict__ x, const float* __restrict__ w,
                               float* __restrict__ out, int Mtot) {
  const int lane = threadIdx.x & 31;
  const int wave = (blockIdx.x * blockDim.x + threadIdx.x) >> 5;
  const int rowBase = wave * 16;
  if (rowBase >= Mtot) return;          // wave-uniform exit: EXEC stays all-ones
  const int lrow = lane & 15;           // N index (also M offset for A loads)
  const int hi = lane >> 4;             // half-wave select
  const int row = rowBase + lrow;       // always < Mtot (Mtot % 16 == 0)
  const float wl[4] = {w[0], w[1], w[2], w[3]};
  v8f acc = {0.f, 0.f, 0.f, 0.f, 0.f, 0.f, 0.f, 0.f};
#pragma unroll
  for (int q = 0; q < 4; ++q) {
    // A 16x4 f32 fragment: lanes 0-15 hold K=0,1 ; lanes 16-31 hold K=2,3
    const int j0 = q * 4 + hi * 2;
    const float2 av = *reinterpret_cast<const float2*>(x + row * 16 + j0);
    v2f a; a.x = av.x; a.y = av.y;
    // B 4x16 fragment (row striped across lanes): this lane holds B[k][lrow], k=hi*2,hi*2+1
    const int kA = hi * 2, kB = hi * 2 + 1;
    v2f bm;
    bm.x = (lrow == kA) ? wl[q] : 0.0f;
    bm.y = (lrow == kB) ? wl[q] : 0.0f;
    acc = __builtin_amdgcn_wmma_f32_16x16x4_f32(false, a, false, bm, (short)0, acc,
                                                false, false);
  }
  // D 16x16 f32: VGPR v holds M=v (lanes 0-15) / M=v+8 (lanes 16-31), N=lane%16.
  // Single predicate for all 8 stores; rows are guaranteed in range.
  if (lrow < Cdim) {
    float* o = out + (rowBase + hi * 8) * Cdim + lrow;
#pragma unroll
    for (int v = 0; v < 8; ++v) {
      o[v * Cdim] = acc[v];
    }
  }
}

// ---------- host orchestration ----------
extern "C" void kernel_launch(void* const* d_in, const int* in_sizes, int n_in,
                              void* d_out, int out_size, void* d_ws, size_t ws_size,
                              hipStream_t stream) {
  (void)in_sizes; (void)n_in; (void)out_size;
  const float* y        = (const float*)d_in[0];
  const float* u_ew     = (const float*)d_in[1];
  const float* d_ew     = (const float*)d_in[2];
  const float* mu_u     = (const float*)d_in[3];
  const float* mu_d1    = (const float*)d_in[4];
  const float* mu_d2    = (const float*)d_in[5];
  const float* rho      = (const float*)d_in[6];
  const float* rho_u    = (const float*)d_in[7];
  const float* rho_d    = (const float*)d_in[8];
  const float* alpha_x  = (const float*)d_in[9];
  const float* beta_x   = (const float*)d_in[10];
  const float* alpha_zu = (const float*)d_in[11];
  const float* beta_zu  = (const float*)d_in[12];
  const float* alpha_zd = (const float*)d_in[13];
  const float* beta_zd  = (const float*)d_in[14];
  const float* comb     = (const float*)d_in[15];
  const int*   knn      = (const int*)d_in[16];
  const int*   maskp    = (const int*)d_in[17];

  if (ws_size < (size_t)13 * NELEM * sizeof(float)) return;  // scratch guard

  float* ws  = (float*)d_ws;
  float* X   = ws + 0  * (size_t)NELEM;
  float* ZU  = ws + 1  * (size_t)NELEM;
  float* ZD  = ws + 2  * (size_t)NELEM;
  float* G   = ws + 3  * (size_t)NELEM;
  float* GU  = ws + 4  * (size_t)NELEM;
  float* GD  = ws + 5  * (size_t)NELEM;
  float* PHI = ws + 6  * (size_t)NELEM;
  float* T1  = ws + 7  * (size_t)NELEM;
  float* T2  = ws + 8  * (size_t)NELEM;
  float* RHS = ws + 9  * (size_t)NELEM;
  float* R   = ws + 10 * (size_t)NELEM;
  float* P   = ws + 11 * (size_t)NELEM;
  float* AP  = ws + 12 * (size_t)NELEM;

  const dim3 blk(256);
  const dim3 grd((NV4 + 255) / 256);

  k_init<<<grd, blk, 0, stream>>>(X, ZU, ZD, G, GU, GD, y);
  k_ldr <<<grd, blk, 0, stream>>>(PHI, X, d_ew, knn);

  for (int i = 0; i < NADMM; ++i) {
    // ---- rhs_x = 0.5*ldrt(gamma + rho*phi) + [0.5*(rho_u*zu + rho_d*zd - gu - gd) + Hty]
    k_t1    <<<grd, blk, 0, stream>>>(T1, G, PHI, rho, i);
    k_rhspre<<<grd, blk, 0, stream>>>(T2, ZU, ZD, GU, GD, y, rho_u, rho_d, i, maskp);
    k_rhsx2 <<<grd, blk, 0, stream>>>(RHS, T1, T2, d_ew, knn);

    // ---- CG on x with A(v) = tmask*v + rho/2*ldrt(ldr(v)) + (rho_u+rho_d)/2*v
    auto Ax = [&](const float* vin) {
      k_ldr  <<<grd, blk, 0, stream>>>(T1, vin, d_ew, knn);
      k_lhsx2<<<grd, blk, 0, stream>>>(AP, T1, vin, d_ew, knn, rho, rho_u, rho_d, i, maskp);
    };
    Ax(X);
    k_residual<<<grd, blk, 0, stream>>>(R, P, RHS, AP);
    for (int j = 0; j < NCG; ++j) {
      Ax(P);
      k_cgupdate<<<grd, blk, 0, stream>>>(X, R, P, AP, alpha_x, beta_x, (i * NCG + j) * Hdim);
    }

    // ---- CG on zu with A(v) = mu_u*lu(v) + rho_u/2*v
    k_halfaxpy<<<grd, blk, 0, stream>>>(RHS, GU, X, rho_u, i);
    k_lu      <<<grd, blk, 0, stream>>>(AP, ZU, u_ew, knn, mu_u, rho_u, i);
    k_residual<<<grd, blk, 0, stream>>>(R, P, RHS, AP);
    for (int j = 0; j < NCG; ++j) {
      k_lu      <<<grd, blk, 0, stream>>>(AP, P, u_ew, knn, mu_u, rho_u, i);
      k_cgupdate<<<grd, blk, 0, stream>>>(ZU, R, P, AP, alpha_zu, beta_zu, (i * NCG + j) * Hdim);
    }

    // ---- CG on zd with A(v) = mu_d2*ldrt(ldr(v)) + rho_d/2*v
    k_halfaxpy<<<grd, blk, 0, stream>>>(RHS, GD, X, rho_d, i);
    auto Azd = [&](const float* vin) {
      k_ldr   <<<grd, blk, 0, stream>>>(T1, vin, d_ew, knn);
      k_lhszd2<<<grd, blk, 0, stream>>>(AP, T1, vin, d_ew, knn, mu_d2, rho_d, i);
    };
    Azd(ZD);
    k_residual<<<grd, blk, 0, stream>>>(R, P, RHS, AP);
    for (int j = 0; j < NCG; ++j) {
      Azd(P);
      k_cgupdate<<<grd, blk, 0, stream>>>(ZD, R, P, AP, alpha_zd, beta_zd, (i * NCG + j) * Hdim);
    }

    // ---- dual + shrinkage updates
    k_gammaupd<<<grd, blk, 0, stream>>>(GU, X, ZU, rho_u, i);
    k_gammaupd<<<grd, blk, 0, stream>>>(GD, X, ZD, rho_d, i);
    k_ldr     <<<grd, blk, 0, stream>>>(T1, X, d_ew, knn);
    k_shrink  <<<grd, blk, 0, stream>>>(PHI, G, T1, rho, mu_d1, i);
  }

  // ---- final combine over H via f32 WMMA (16 rows per wave, 8 waves per block)
  const dim3 cgrd((MROWS + 127) / 128);
  k_combine_wmma<<<cgrd, dim3(256), 0, stream>>>(X, comb, (float*)d_out, MROWS);
}